// AttentionWeightedAverage_47811575939243
// MI455X (gfx1250) — compile-verified
//
#include <hip/hip_runtime.h>
#include <math.h>

// ---------------------------------------------------------------------------
// AttentionWeightedAverage for MI455X (gfx1250, wave32, WMMA).
// Shapes: h(512,1) v(384,384) wh(384,512) wv(512,384) wg(512,512)
// out(384,384,384) f32  -> 226 MB write-dominated; GEMMs done with
// v_wmma_f32_16x16x4_f32 (exact f32 path), final stream uses NT b128 stores.
// ---------------------------------------------------------------------------

typedef __attribute__((ext_vector_type(2))) float v2f;
typedef __attribute__((ext_vector_type(4))) float f4;
typedef __attribute__((ext_vector_type(8))) float v8f;

#define SEQDIM   512
#define EMBED    512
#define VDIM     384
#define VLEN     384

// ---------------- Kernel 1: gh = wg @ h  (512-vector) ----------------------
__global__ void k_matvec(const float* __restrict__ wg,
                         const float* __restrict__ h,
                         float* __restrict__ gh) {
    int e = blockIdx.x * blockDim.x + threadIdx.x;
    if (e < EMBED) {
        const float* row = wg + e * SEQDIM;
        float acc = 0.0f;
#pragma unroll 8
        for (int t = 0; t < SEQDIM; ++t) acc = fmaf(row[t], h[t], acc);
        gh[e] = acc;
    }
}

// ------------- Kernel 2: t = tanh(wv @ v + gh broadcast) -------------------
// A = wv (512 x 384), B = v (384 x 384), C/D 16x16 tile per wave.
// WMMA f32 16x16x4 fragment layout (ISA 7.12.2):
//   A: lanes 0-15 row m0+l16 K={kk,kk+1}; lanes 16-31 K={kk+2,kk+3}
//   B: lanes 0-15 col n0+l16 K={kk,kk+1}; lanes 16-31 K={kk+2,kk+3}
//   C/D vgpr j: lanes 0-15 -> row m0+j, lanes 16-31 -> row m0+8+j
__global__ void k_s_tanh(const float* __restrict__ wv,
                         const float* __restrict__ v,
                         const float* __restrict__ gh,
                         float* __restrict__ t) {
    const int M = EMBED, K = VDIM, N = VLEN;
    const int nTiles = N / 16;                       // 24
    int wave = (blockIdx.x * blockDim.x + threadIdx.x) >> 5;
    int lane = threadIdx.x & 31;
    int half = lane >> 4;
    int l16  = lane & 15;
    int m0 = (wave / nTiles) * 16;
    int n0 = (wave % nTiles) * 16;
    if (m0 >= M) return;

    v8f acc;
#pragma unroll
    for (int j = 0; j < 8; ++j) acc[j] = gh[m0 + j + 8 * half];  // + (Wg h) 1^T

    const float* ap = wv + (m0 + l16) * K + 2 * half;
    const float* bp = v  + (2 * half) * N + n0 + l16;
    for (int kk = 0; kk < K; kk += 4) {
        v2f a, b;
        a.x = ap[kk];
        a.y = ap[kk + 1];
        b.x = bp[kk * N];
        b.y = bp[kk * N + N];
        acc = __builtin_amdgcn_wmma_f32_16x16x4_f32(
                  false, a, false, b, (short)0, acc, false, false);
    }
#pragma unroll
    for (int j = 0; j < 8; ++j) {
        int row = m0 + j + 8 * half;
        t[row * N + n0 + l16] = tanhf(acc[j]);
    }
}

// ------------------- Kernel 3: z = wh @ t  (384 x 384) ---------------------
__global__ void k_z(const float* __restrict__ wh,
                    const float* __restrict__ t,
                    float* __restrict__ z) {
    const int M = VLEN, K = EMBED, N = VLEN;
    const int nTiles = N / 16;                       // 24
    int wave = (blockIdx.x * blockDim.x + threadIdx.x) >> 5;
    int lane = threadIdx.x & 31;
    int half = lane >> 4;
    int l16  = lane & 15;
    int m0 = (wave / nTiles) * 16;
    int n0 = (wave % nTiles) * 16;
    if (m0 >= M) return;

    v8f acc = {};
    const float* ap = wh + (m0 + l16) * K + 2 * half;
    const float* bp = t  + (2 * half) * N + n0 + l16;
    for (int kk = 0; kk < K; kk += 4) {
        v2f a, b;
        a.x = ap[kk];
        a.y = ap[kk + 1];
        b.x = bp[kk * N];
        b.y = bp[kk * N + N];
        acc = __builtin_amdgcn_wmma_f32_16x16x4_f32(
                  false, a, false, b, (short)0, acc, false, false);
    }
#pragma unroll
    for (int j = 0; j < 8; ++j) {
        int row = m0 + j + 8 * half;
        z[row * N + n0 + l16] = acc[j];
    }
}

// ---------------- Kernel 4: alpha = softmax(z, axis=-1) --------------------
// One 128-thread block per row; 3 elements per thread (384 = 3*128).
__global__ void k_softmax(const float* __restrict__ z,
                          float* __restrict__ alpha) {
    __shared__ float red[128];
    int row = blockIdx.x;
    int tid = threadIdx.x;
    const float* zr = z + row * VLEN;

    float x0 = zr[tid], x1 = zr[tid + 128], x2 = zr[tid + 256];
    red[tid] = fmaxf(fmaxf(x0, x1), x2);
    __syncthreads();
    for (int s = 64; s > 0; s >>= 1) {
        if (tid < s) red[tid] = fmaxf(red[tid], red[tid + s]);
        __syncthreads();
    }
    float m = red[0];
    __syncthreads();

    float e0 = __expf(x0 - m), e1 = __expf(x1 - m), e2 = __expf(x2 - m);
    red[tid] = e0 + e1 + e2;
    __syncthreads();
    for (int s = 64; s > 0; s >>= 1) {
        if (tid < s) red[tid] += red[tid + s];
        __syncthreads();
    }
    float inv = 1.0f / red[0];

    float* ar = alpha + row * VLEN;
    ar[tid]       = e0 * inv;
    ar[tid + 128] = e1 * inv;
    ar[tid + 256] = e2 * inv;
}

// -------- Kernel 5: out[i][d][l] = v[d][l] * alpha[i][l]  (226 MB) ---------
// One float4 per thread; v/alpha stay L2-resident, output streamed with
// non-temporal b128 stores so write-once data doesn't pollute caches.
__global__ void k_weighted(const float* __restrict__ v,
                           const float* __restrict__ alpha,
                           float* __restrict__ out) {
    const int LQ = VLEN / 4;                 // 96 float4 per row
    int idx = blockIdx.x * blockDim.x + threadIdx.x;   // total = 384*384*96
    int i   = idx / (VDIM * LQ);
    int rem = idx - i * (VDIM * LQ);
    int d   = rem / LQ;
    int lq  = rem - d * LQ;

    const f4* v4 = (const f4*)v;
    const f4* a4 = (const f4*)alpha;
    f4 r = v4[d * LQ + lq] * a4[i * LQ + lq];
    __builtin_nontemporal_store(r, (f4*)out + idx);
}

// ---------------------------------------------------------------------------
extern "C" void kernel_launch(void* const* d_in, const int* in_sizes, int n_in,
                              void* d_out, int out_size, void* d_ws, size_t ws_size,
                              hipStream_t stream) {
    const float* h  = (const float*)d_in[0];   // 512
    const float* v  = (const float*)d_in[1];   // 384*384
    const float* wh = (const float*)d_in[2];   // 384*512
    const float* wv = (const float*)d_in[3];   // 512*384
    const float* wg = (const float*)d_in[4];   // 512*512
    float* out = (float*)d_out;                // 384*384*384

    float* ws    = (float*)d_ws;
    float* gh    = ws;                               // 512
    float* t     = gh + EMBED;                       // 512*384
    float* z     = t + EMBED * VLEN;                 // 384*384
    float* alpha = z + VLEN * VLEN;                  // 384*384

    // 1) gh = wg @ h
    k_matvec<<<2, 256, 0, stream>>>(wg, h, gh);

    // 2) t = tanh(wv @ v + gh)   : 32x24 = 768 tiles, 8 waves/block -> 96 blocks
    k_s_tanh<<<96, 256, 0, stream>>>(wv, v, gh, t);

    // 3) z = wh @ t              : 24x24 = 576 tiles -> 72 blocks
    k_z<<<72, 256, 0, stream>>>(wh, t, z);

    // 4) alpha = softmax rows
    k_softmax<<<VLEN, 128, 0, stream>>>(z, alpha);

    // 5) out = v * alpha[..., None]  (384^3 floats = 14,155,776 float4)
    k_weighted<<<(VLEN * VDIM * (VLEN / 4)) / 256, 256, 0, stream>>>(v, alpha, out);
}